// DualGNN_Bilinear_2362232013505
// MI455X (gfx1250) — compile-verified
//
#include <hip/hip_runtime.h>
#include <hip/hip_bf16.h>
#include <math.h>

// Problem constants (match reference)
#define NBG   1024      // graphs
#define NLIG  65536     // ligand nodes
#define NPRO  262144    // protein nodes
#define ELIG  524288    // ligand edges
#define EPRO  2097152   // protein edges

typedef __attribute__((ext_vector_type(2))) float v2f;
typedef __attribute__((ext_vector_type(8))) float v8f;

// ---------------------------------------------------------------------------
// Elementwise helpers
// ---------------------------------------------------------------------------
__global__ __launch_bounds__(256) void fill_f32_kernel(float* __restrict__ p,
                                                       long long n, float v) {
  long long t = (long long)blockIdx.x * 256 + threadIdx.x;
  if (t < n) p[t] = v;
}

__global__ __launch_bounds__(256) void deg_kernel(const int* __restrict__ dst,
                                                  float* __restrict__ deg, int E) {
  int t = blockIdx.x * 256 + threadIdx.x;
  if (t < E) atomicAdd(&deg[dst[t]], 1.0f);
}

__global__ __launch_bounds__(256) void rsqrt_kernel(float* __restrict__ d, int n) {
  int t = blockIdx.x * 256 + threadIdx.x;
  if (t < n) d[t] = rsqrtf(d[t]);   // deg >= 1 always (self loop)
}

// ---------------------------------------------------------------------------
// WMMA f32 16x16x4 tiled matmul: Out[M,N] = act(X[M,K] @ W[K,N] + bias)
// blockDim = (32,4): 4 waves/block, each wave owns one 16x16 output tile.
// Main K-loop is guard-free (full-EXEC b64 A-loads + coalesced B-loads);
// the K%4 tail (78->2, 30->2) is a single guarded iteration.
// A-frag  : lane l (hi=l>>4): row = tm*16+(l&15), K = {k+2*hi, k+2*hi+1}
// B-frag  : lane l:           col = tn*16+(l&15), same K split
// C/D     : VGPR r -> row M = r + 8*hi   (CDNA5 ISA layouts)
// ---------------------------------------------------------------------------
__global__ __launch_bounds__(128)
void wmma_mm_kernel(const float* __restrict__ X, const float* __restrict__ W,
                    const float* __restrict__ bias, float* __restrict__ Out,
                    int M, int K, int N, int act) {
  const int lane = threadIdx.x;
  const int l16  = lane & 15;
  const int hi   = lane >> 4;
  const int tn   = blockIdx.x;
  const int tm   = blockIdx.y * 4 + threadIdx.y;
  const int row  = tm * 16 + l16;      // A row owned by this lane
  const int col  = tn * 16 + l16;      // B/C column owned by this lane
  const int kofs = hi * 2;

  const float* Xr = X + (size_t)row * K + kofs;        // A: X[row][k+kofs]
  const float* Wc = W + (size_t)kofs * N + col;        // B: W[k+kofs][col]
  const size_t Wstep = (size_t)N;                      // one K row of W

  v8f acc = {};
  const int Kmain = K & ~3;
  int k = 0;
#pragma unroll 2
  for (; k < Kmain; k += 4) {
    // contiguous K-pair per lane -> single global_load_b64
    v2f a = *(const v2f*)(Xr + k);
    v2f b;
    const float* wp = Wc + (size_t)k * Wstep;
    b.x = wp[0];
    b.y = wp[Wstep];
    acc = __builtin_amdgcn_wmma_f32_16x16x4_f32(
        false, a, false, b, (short)0, acc, false, false);
  }
  if (k < K) {  // single guarded tail iteration (K % 4 != 0)
    v2f a = {0.0f, 0.0f};
    v2f b = {0.0f, 0.0f};
    const float* wp = Wc + (size_t)k * Wstep;
    if (k + kofs < K)     { a.x = Xr[k];     b.x = wp[0]; }
    if (k + kofs + 1 < K) { a.y = Xr[k + 1]; b.y = wp[Wstep]; }
    acc = __builtin_amdgcn_wmma_f32_16x16x4_f32(
        false, a, false, b, (short)0, acc, false, false);
  }

  const float bv = bias ? bias[col] : 0.0f;
#pragma unroll
  for (int r = 0; r < 8; ++r) {
    const int orow = tm * 16 + r + hi * 8;
    float v = acc[r] + bv;
    if (act == 1) v = fmaxf(v, 0.0f);
    Out[(size_t)orow * N + col] = v;
  }
}

// ---------------------------------------------------------------------------
// Edge scatter-add: out[dst] += h[src] * dinv[src]*dinv[dst]
// Thread = (edge, feature); F = 1<<Fs (32 or 64) so each wave stays within one
// feature row -> coalesced reads of h[src] and coalesced atomics on out[dst].
// ---------------------------------------------------------------------------
__global__ __launch_bounds__(256)
void edge_scatter_kernel(const int* __restrict__ ei, long long E,
                         const float* __restrict__ h,
                         const float* __restrict__ dinv,
                         float* __restrict__ out, int Fs) {
  const long long total = E << Fs;
  long long t = (long long)blockIdx.x * 256 + threadIdx.x;
  if (t >= total) return;
  const long long e = t >> Fs;
  const int f = (int)(t & ((1 << Fs) - 1));
  const int src = ei[e];
  const int dst = ei[E + e];
  const float nrm = dinv[src] * dinv[dst];
  atomicAdd(&out[((size_t)dst << Fs) + f], h[((size_t)src << Fs) + f] * nrm);
}

// out = relu(out + h*dinv^2 + bias)   (self-loop term + bias + activation)
__global__ __launch_bounds__(256)
void gcn_finalize_kernel(float* __restrict__ out, const float* __restrict__ h,
                         const float* __restrict__ dinv,
                         const float* __restrict__ bias, long long n, int Fs) {
  long long t = (long long)blockIdx.x * 256 + threadIdx.x;
  if (t >= (n << Fs)) return;
  const long long i = t >> Fs;
  const int f = (int)(t & ((1 << Fs) - 1));
  const float di = dinv[i];
  out[t] = fmaxf(out[t] + h[t] * di * di + bias[f], 0.0f);
}

// ---------------------------------------------------------------------------
// Mean pool (F = 32)
// ---------------------------------------------------------------------------
__global__ __launch_bounds__(256)
void pool_accum_kernel(const float* __restrict__ x, const int* __restrict__ batch,
                       float* __restrict__ pool, float* __restrict__ cnt,
                       long long n) {
  long long t = (long long)blockIdx.x * 256 + threadIdx.x;
  if (t >= (n << 5)) return;
  const long long i = t >> 5;
  const int f = (int)(t & 31);
  const int g = batch[i];
  atomicAdd(&pool[(size_t)g * 32 + f], x[t]);
  if (f == 0) atomicAdd(&cnt[g], 1.0f);
}

__global__ __launch_bounds__(256)
void pool_div_kernel(float* __restrict__ pool, const float* __restrict__ cnt,
                     int total) {
  int t = blockIdx.x * 256 + threadIdx.x;
  if (t < total) pool[t] /= fmaxf(cnt[t >> 5], 1.0f);
}

// ---------------------------------------------------------------------------
// Bilinear attention core: bil[b,h] = relu( xl[b]^T Wb[h] xp[b] + bb[h] )
// ---------------------------------------------------------------------------
__global__ __launch_bounds__(256)
void bilinear_kernel(const float* __restrict__ xl, const float* __restrict__ xp,
                     const float* __restrict__ Wb, const float* __restrict__ bb,
                     float* __restrict__ bil) {
  int t = blockIdx.x * 256 + threadIdx.x;
  if (t >= NBG * 128) return;
  const int b = t >> 7;
  const int h = t & 127;
  const float* xlr = xl + (size_t)b * 32;
  const float* xpr = xp + (size_t)b * 32;
  const float* Wh  = Wb + (size_t)h * 1024;
  float acc = 0.0f;
#pragma unroll 4
  for (int i = 0; i < 32; ++i) {
    const float xi = xlr[i];
    const float* Wr = Wh + i * 32;
    for (int j = 0; j < 32; ++j) acc = fmaf(xi * Wr[j], xpr[j], acc);
  }
  bil[t] = fmaxf(acc + bb[h], 0.0f);
}

// out[b] = sigmoid( dot(X[b,0:K], w) + c[0] )
__global__ __launch_bounds__(256)
void dot_sigmoid_kernel(const float* __restrict__ X, const float* __restrict__ w,
                        const float* __restrict__ c, float* __restrict__ out,
                        int rows, int K) {
  int b = blockIdx.x * 256 + threadIdx.x;
  if (b >= rows) return;
  float acc = c[0];
  const float* xr = X + (size_t)b * K;
  for (int k = 0; k < K; ++k) acc = fmaf(xr[k], w[k], acc);
  out[b] = 1.0f / (1.0f + expf(-acc));
}

// attended[b,h] = attw[b] * bil[b,h]
__global__ __launch_bounds__(256)
void attend_kernel(float* __restrict__ att, const float* __restrict__ bil,
                   const float* __restrict__ attw, int total) {
  int t = blockIdx.x * 256 + threadIdx.x;
  if (t < total) att[t] = attw[t >> 7] * bil[t];
}

// ---------------------------------------------------------------------------
// Launch
// ---------------------------------------------------------------------------
static inline int cdiv(long long a, long long b) { return (int)((a + b - 1) / b); }

extern "C" void kernel_launch(void* const* d_in, const int* in_sizes, int n_in,
                              void* d_out, int out_size, void* d_ws, size_t ws_size,
                              hipStream_t stream) {
  // ---- inputs (setup_inputs() order) ----
  const float* ligand_x  = (const float*)d_in[0];
  const float* protein_x = (const float*)d_in[1];
  const int*   lei       = (const int*)d_in[2];   // [2, ELIG] flat
  const int*   pei       = (const int*)d_in[3];   // [2, EPRO] flat
  const int*   lbatch    = (const int*)d_in[4];
  const int*   pbatch    = (const int*)d_in[5];
  const float* lW1 = (const float*)d_in[6];  const float* lb1 = (const float*)d_in[7];
  const float* lW2 = (const float*)d_in[8];  const float* lb2 = (const float*)d_in[9];
  const float* pW1 = (const float*)d_in[10]; const float* pb1 = (const float*)d_in[11];
  const float* pW2 = (const float*)d_in[12]; const float* pb2 = (const float*)d_in[13];
  const float* Wb  = (const float*)d_in[14]; const float* bb  = (const float*)d_in[15];
  const float* A1  = (const float*)d_in[16]; const float* a1  = (const float*)d_in[17];
  const float* A2  = (const float*)d_in[18]; const float* a2  = (const float*)d_in[19];
  const float* O1  = (const float*)d_in[20]; const float* o1  = (const float*)d_in[21];
  const float* O2  = (const float*)d_in[22]; const float* o2  = (const float*)d_in[23];
  const float* F1  = (const float*)d_in[24]; const float* f1  = (const float*)d_in[25];
  const float* F2  = (const float*)d_in[26]; const float* f2  = (const float*)d_in[27];
  const float* F3  = (const float*)d_in[28]; const float* f3  = (const float*)d_in[29];
  float* out = (float*)d_out;

  // ---- workspace carve-up (256B aligned) ----
  char* base = (char*)d_ws;
  size_t off = 0;
  auto carve = [&](size_t bytes) -> float* {
    float* p = (float*)(base + off);
    off = (off + bytes + 255) & ~(size_t)255;
    return p;
  };
  float* bufH  = carve((size_t)NPRO * 64 * 4);   // pre-aggregation h
  float* bufA  = carve((size_t)NPRO * 64 * 4);   // aggregated / activated
  float* dinvL = carve((size_t)NLIG * 4);
  float* dinvP = carve((size_t)NPRO * 4);
  float* poolL = carve((size_t)NBG * 32 * 4);
  float* poolP = carve((size_t)NBG * 32 * 4);
  float* cnt   = carve((size_t)NBG * 4);
  float* bil   = carve((size_t)NBG * 128 * 4);
  float* t1    = carve((size_t)NBG * 128 * 4);
  float* t2    = carve((size_t)NBG * 128 * 4);
  float* attw  = carve((size_t)NBG * 4);
  float* att   = carve((size_t)NBG * 128 * 4);
  float* feat  = carve((size_t)NBG * 64 * 4);
  (void)ws_size; (void)in_sizes; (void)n_in; (void)out_size;

  const dim3 blk256(256);
  const dim3 wblk(32, 4);   // 4 waves per block, one 16x16 tile each

  // ================= Ligand GNN =================
  fill_f32_kernel<<<cdiv(NLIG, 256), blk256, 0, stream>>>(dinvL, NLIG, 1.0f);
  deg_kernel<<<cdiv(ELIG, 256), blk256, 0, stream>>>(lei + ELIG, dinvL, ELIG);
  rsqrt_kernel<<<cdiv(NLIG, 256), blk256, 0, stream>>>(dinvL, NLIG);

  // layer 1: h = x @ lW1  (K=78, tail=2), aggregate, +b, relu
  wmma_mm_kernel<<<dim3(64 / 16, NLIG / 64), wblk, 0, stream>>>(
      ligand_x, lW1, nullptr, bufH, NLIG, 78, 64, 0);
  fill_f32_kernel<<<cdiv((long long)NLIG * 64, 256), blk256, 0, stream>>>(bufA, (long long)NLIG * 64, 0.0f);
  edge_scatter_kernel<<<cdiv((long long)ELIG << 6, 256), blk256, 0, stream>>>(lei, ELIG, bufH, dinvL, bufA, 6);
  gcn_finalize_kernel<<<cdiv((long long)NLIG << 6, 256), blk256, 0, stream>>>(bufA, bufH, dinvL, lb1, NLIG, 6);

  // layer 2: h = a1 @ lW2 (K=64, N=32)
  wmma_mm_kernel<<<dim3(32 / 16, NLIG / 64), wblk, 0, stream>>>(
      bufA, lW2, nullptr, bufH, NLIG, 64, 32, 0);
  fill_f32_kernel<<<cdiv((long long)NLIG * 32, 256), blk256, 0, stream>>>(bufA, (long long)NLIG * 32, 0.0f);
  edge_scatter_kernel<<<cdiv((long long)ELIG << 5, 256), blk256, 0, stream>>>(lei, ELIG, bufH, dinvL, bufA, 5);
  gcn_finalize_kernel<<<cdiv((long long)NLIG << 5, 256), blk256, 0, stream>>>(bufA, bufH, dinvL, lb2, NLIG, 5);

  // mean pool -> poolL [B,32]
  fill_f32_kernel<<<cdiv(NBG * 32, 256), blk256, 0, stream>>>(poolL, NBG * 32, 0.0f);
  fill_f32_kernel<<<cdiv(NBG, 256), blk256, 0, stream>>>(cnt, NBG, 0.0f);
  pool_accum_kernel<<<cdiv((long long)NLIG << 5, 256), blk256, 0, stream>>>(bufA, lbatch, poolL, cnt, NLIG);
  pool_div_kernel<<<cdiv(NBG * 32, 256), blk256, 0, stream>>>(poolL, cnt, NBG * 32);

  // ================= Protein GNN =================
  fill_f32_kernel<<<cdiv(NPRO, 256), blk256, 0, stream>>>(dinvP, NPRO, 1.0f);
  deg_kernel<<<cdiv(EPRO, 256), blk256, 0, stream>>>(pei + EPRO, dinvP, EPRO);
  rsqrt_kernel<<<cdiv(NPRO, 256), blk256, 0, stream>>>(dinvP, NPRO);

  wmma_mm_kernel<<<dim3(64 / 16, NPRO / 64), wblk, 0, stream>>>(
      protein_x, pW1, nullptr, bufH, NPRO, 30, 64, 0);
  fill_f32_kernel<<<cdiv((long long)NPRO * 64, 256), blk256, 0, stream>>>(bufA, (long long)NPRO * 64, 0.0f);
  edge_scatter_kernel<<<cdiv((long long)EPRO << 6, 256), blk256, 0, stream>>>(pei, EPRO, bufH, dinvP, bufA, 6);
  gcn_finalize_kernel<<<cdiv((long long)NPRO << 6, 256), blk256, 0, stream>>>(bufA, bufH, dinvP, pb1, NPRO, 6);

  wmma_mm_kernel<<<dim3(32 / 16, NPRO / 64), wblk, 0, stream>>>(
      bufA, pW2, nullptr, bufH, NPRO, 64, 32, 0);
  fill_f32_kernel<<<cdiv((long long)NPRO * 32, 256), blk256, 0, stream>>>(bufA, (long long)NPRO * 32, 0.0f);
  edge_scatter_kernel<<<cdiv((long long)EPRO << 5, 256), blk256, 0, stream>>>(pei, EPRO, bufH, dinvP, bufA, 5);
  gcn_finalize_kernel<<<cdiv((long long)NPRO << 5, 256), blk256, 0, stream>>>(bufA, bufH, dinvP, pb2, NPRO, 5);

  fill_f32_kernel<<<cdiv(NBG * 32, 256), blk256, 0, stream>>>(poolP, NBG * 32, 0.0f);
  fill_f32_kernel<<<cdiv(NBG, 256), blk256, 0, stream>>>(cnt, NBG, 0.0f);
  pool_accum_kernel<<<cdiv((long long)NPRO << 5, 256), blk256, 0, stream>>>(bufA, pbatch, poolP, cnt, NPRO);
  pool_div_kernel<<<cdiv(NBG * 32, 256), blk256, 0, stream>>>(poolP, cnt, NBG * 32);

  // ================= Bilinear attention + head =================
  bilinear_kernel<<<cdiv(NBG * 128, 256), blk256, 0, stream>>>(poolL, poolP, Wb, bb, bil);

  // t1 = relu(bil @ A1 + a1)  [B,64]
  wmma_mm_kernel<<<dim3(64 / 16, NBG / 64), wblk, 0, stream>>>(bil, A1, a1, t1, NBG, 128, 64, 1);
  // attw = sigmoid(t1 @ A2 + a2)  [B]
  dot_sigmoid_kernel<<<cdiv(NBG, 256), blk256, 0, stream>>>(t1, A2, a2, attw, NBG, 64);
  // attended = attw * bil  [B,128]
  attend_kernel<<<cdiv(NBG * 128, 256), blk256, 0, stream>>>(att, bil, attw, NBG * 128);
  // t2 = relu(att @ O1 + o1) [B,128]
  wmma_mm_kernel<<<dim3(128 / 16, NBG / 64), wblk, 0, stream>>>(att, O1, o1, t2, NBG, 128, 128, 1);
  // feat = t2 @ O2 + o2 [B,64]  (no relu)
  wmma_mm_kernel<<<dim3(64 / 16, NBG / 64), wblk, 0, stream>>>(t2, O2, o2, feat, NBG, 128, 64, 0);
  // t2 = relu(feat @ F1 + f1) [B,128]
  wmma_mm_kernel<<<dim3(128 / 16, NBG / 64), wblk, 0, stream>>>(feat, F1, f1, t2, NBG, 64, 128, 1);
  // t1 = relu(t2 @ F2 + f2) [B,64]
  wmma_mm_kernel<<<dim3(64 / 16, NBG / 64), wblk, 0, stream>>>(t2, F2, f2, t1, NBG, 128, 64, 1);
  // out = sigmoid(t1 @ F3 + f3) [B,1]
  dot_sigmoid_kernel<<<cdiv(NBG, 256), blk256, 0, stream>>>(t1, F3, f3, out, NBG, 64);
}